// SubspaceLinopFactory_80204219285860
// MI455X (gfx1250) — compile-verified
//
#include <hip/hip_runtime.h>
#include <math.h>

typedef __attribute__((ext_vector_type(2))) float v2f;
typedef __attribute__((ext_vector_type(8))) float v8f;

namespace {
constexpr int A_ = 4, T_ = 256, C_ = 4, H_ = 64, W_ = 64, R_ = 8, K_ = 512;
constexpr int N_  = H_ * W_;   // 4096
constexpr int AC_ = A_ * C_;   // 16 -> exactly the WMMA N dimension
constexpr int KT_ = K_ / 16;   // 32 k-tiles of 16 rows
constexpr int NSPLIT = 4;      // split the n-reduction for occupancy

// workspace layout, in float2 (complex) units
constexpr size_t OFF_XS = 0;                                    // [AC][N]
constexpr size_t OFF_EX = OFF_XS + (size_t)AC_ * N_;            // [R][K][H]
constexpr size_t OFF_EY = OFF_EX + (size_t)R_ * K_ * H_;        // [R][K][W]
constexpr size_t OFF_YP = OFF_EY + (size_t)R_ * K_ * W_;        // [NSPLIT][A][R][C][K]
} // namespace

// ---------------------------------------------------------------------------
// Kernel 1: separable phase tables  Ex[r,k,i] = exp(-i*trj[r,0,k]*(i-32)),
//                                   Ey[r,k,j] = exp(-i*trj[r,1,k]*(j-32)).
// Only 0.5M sincos total instead of 16.7M in the inner GEMM.
// ---------------------------------------------------------------------------
__global__ void nudft_tables(const float* __restrict__ trj, float2* __restrict__ ws)
{
    int idx = blockIdx.x * blockDim.x + threadIdx.x;
    const int per = R_ * K_ * 64;            // 262144 entries per table
    int d   = (idx >= per) ? 1 : 0;          // 0 -> Ex (H axis), 1 -> Ey (W axis)
    int rem = idx - d * per;                 // rem = (r*K + k)*64 + g
    int g = rem & 63;
    int k = (rem >> 6) & (K_ - 1);
    int r = rem >> 15;
    float t  = trj[(r * 2 + d) * K_ + k];
    float ph = t * (float)(g - 32);
    float s, c;
    sincosf(ph, &s, &c);
    ws[(d ? OFF_EY : OFF_EX) + (size_t)rem] = make_float2(c, -s);
}

// ---------------------------------------------------------------------------
// Kernel 2: xs[(a,c), n] = x[a,n] * mps[c,n]   (complex), the 16-column B panel
// ---------------------------------------------------------------------------
__global__ void build_xs(const float2* __restrict__ x, const float2* __restrict__ mps,
                         float2* __restrict__ ws)
{
    int idx = blockIdx.x * blockDim.x + threadIdx.x;   // idx = ((a*C + c)*N) + n
    int n = idx & (N_ - 1);
    int c = (idx >> 12) & (C_ - 1);
    int a = idx >> 14;
    float2 xv = x[a * N_ + n];
    float2 mv = mps[c * N_ + n];
    ws[OFF_XS + (size_t)idx] =
        make_float2(xv.x * mv.x - xv.y * mv.y, xv.x * mv.y + xv.y * mv.x);
}

// ---------------------------------------------------------------------------
// Kernel 3: the NUDFT GEMM on WMMA f32 16x16x4.
// Block = (r, k-tile, n-split); 8 wave32 per block, each wave owns 128 n,
// 4 accumulators (Are*Bre, Are*Bim, Aim*Bre, Aim*Bim), LDS cross-wave reduce.
// ---------------------------------------------------------------------------
__global__ __launch_bounds__(256)
void nudft_wmma(float2* __restrict__ ws)
{
    __shared__ float red[8 * 4 * 8 * 32];   // [wave][acc][vgpr][lane] = 32 KB

    const int b  = blockIdx.x;
    const int ns = b & (NSPLIT - 1);
    const int kt = (b >> 2) & (KT_ - 1);
    const int r  = b >> 7;

    const int tid  = threadIdx.x;
    const int wave = tid >> 5;
    const int lane = tid & 31;
    const int low  = lane & 15;
    const int hi   = lane >> 4;             // hi half of wave -> n-offset +2

    const float2* XS = ws + OFF_XS;
    const float2* EX = ws + OFF_EX;
    const float2* EY = ws + OFF_EY;

    // A-matrix row for this lane is k-local = low; B-matrix column is ac = low.
    const int kk = kt * 16 + low;
    const float2* exRow = EX + (size_t)(r * K_ + kk) * H_;
    const float2* eyRow = EY + (size_t)(r * K_ + kk) * W_;
    const float2* xsRow = XS + (size_t)low * N_;

    v8f accRR = {}, accRI = {}, accIR = {}, accII = {};

    const int nstart = ns * 1024 + wave * 128;
    for (int nb = nstart; nb < nstart + 128; nb += 4) {
        // Per ISA A 16x4 layout: reg0 holds n = nb + 2*hi, reg1 holds n+1.
        // B 4x16 uses the identical per-lane n mapping (cols striped on lanes).
        const int n0 = nb + 2 * hi;
        const int i  = n0 >> 6;             // image row (shared by the 4-chunk)
        const int j  = n0 & 63;             // image col (j+1 never wraps: chunk aligned)

        float2 ex  = exRow[i];
        float2 ey0 = eyRow[j];
        float2 ey1 = eyRow[j + 1];

        v2f are, aim, bre, bim;
        are[0] = ex.x * ey0.x - ex.y * ey0.y;   // Re(Ex*Ey)
        aim[0] = ex.x * ey0.y + ex.y * ey0.x;   // Im(Ex*Ey)
        are[1] = ex.x * ey1.x - ex.y * ey1.y;
        aim[1] = ex.x * ey1.y + ex.y * ey1.x;

        float2 b0 = xsRow[n0];
        float2 b1 = xsRow[n0 + 1];
        bre[0] = b0.x; bre[1] = b1.x;
        bim[0] = b0.y; bim[1] = b1.y;

#if __has_builtin(__builtin_amdgcn_wmma_f32_16x16x4_f32)
        accRR = __builtin_amdgcn_wmma_f32_16x16x4_f32(false, are, false, bre, (short)0, accRR, false, false);
        accRI = __builtin_amdgcn_wmma_f32_16x16x4_f32(false, are, false, bim, (short)0, accRI, false, false);
        accIR = __builtin_amdgcn_wmma_f32_16x16x4_f32(false, aim, false, bre, (short)0, accIR, false, false);
        accII = __builtin_amdgcn_wmma_f32_16x16x4_f32(false, aim, false, bim, (short)0, accII, false, false);
#else
        // compile-keepalive fallback (histogram will show wmma==0 if taken)
        accRR[0] += are[0] * bre[0]; accRI[0] += are[0] * bim[0];
        accIR[0] += aim[0] * bre[0]; accII[0] += aim[0] * bim[0];
#endif
    }

    // spill accumulators to LDS: [wave][acc q][vgpr v][lane]
    float* rp = red + (size_t)wave * 4 * 8 * 32;
    #pragma unroll
    for (int v = 0; v < 8; ++v) {
        rp[(0 * 8 + v) * 32 + lane] = accRR[v];
        rp[(1 * 8 + v) * 32 + lane] = accRI[v];
        rp[(2 * 8 + v) * 32 + lane] = accIR[v];
        rp[(3 * 8 + v) * 32 + lane] = accII[v];
    }
    __syncthreads();

    // Deterministic cross-wave reduce; thread tid owns D element (m, col).
    // D layout: VGPR v, lane L -> row M = v + 8*(L>=16), col = L&15.
    const int m   = tid >> 4;
    const int col = tid & 15;
    const int v   = m & 7;
    const int l   = ((m & 8) ? 16 : 0) | col;
    float rr = 0.f, ri = 0.f, ir = 0.f, ii = 0.f;
    #pragma unroll
    for (int w = 0; w < 8; ++w) {
        const float* q = red + (size_t)w * 4 * 8 * 32;
        rr += q[(0 * 8 + v) * 32 + l];
        ri += q[(1 * 8 + v) * 32 + l];
        ir += q[(2 * 8 + v) * 32 + l];
        ii += q[(3 * 8 + v) * 32 + l];
    }
    const int a = col >> 2, c = col & 3;
    const int k = kt * 16 + m;
    ws[OFF_YP + ((((size_t)ns * A_ + a) * R_ + r) * C_ + c) * K_ + k] =
        make_float2(rr - ii, ri + ir);   // complex combine
}

// ---------------------------------------------------------------------------
// Kernel 4: out[t,c,k] = sqrt_dcf[sub[t],k] * sum_a phi[a,t] * Y[a,sub[t],c,k]
// (also folds the NSPLIT partial sums)
// ---------------------------------------------------------------------------
__global__ void project_out(const float* __restrict__ phi, const float* __restrict__ sqrt_dcf,
                            const int* __restrict__ subsamp, const float2* __restrict__ ws,
                            float2* __restrict__ out)
{
    int idx = blockIdx.x * blockDim.x + threadIdx.x;   // [t][c][k]
    int k = idx & (K_ - 1);
    int c = (idx >> 9) & (C_ - 1);
    int t = idx >> 11;
    int rI = subsamp[t];
    const float2* YP = ws + OFF_YP;
    float yre = 0.f, yim = 0.f;
    #pragma unroll
    for (int a = 0; a < A_; ++a) {
        float sre = 0.f, sim = 0.f;
        #pragma unroll
        for (int nsp = 0; nsp < NSPLIT; ++nsp) {
            float2 yp = YP[((((size_t)nsp * A_ + a) * R_ + rI) * C_ + c) * K_ + k];
            sre += yp.x; sim += yp.y;
        }
        float p = phi[a * T_ + t];
        yre += p * sre;
        yim += p * sim;
    }
    float sc = sqrt_dcf[rI * K_ + k];
    out[idx] = make_float2(yre * sc, yim * sc);
}

// ---------------------------------------------------------------------------
extern "C" void kernel_launch(void* const* d_in, const int* in_sizes, int n_in,
                              void* d_out, int out_size, void* d_ws, size_t ws_size,
                              hipStream_t stream)
{
    (void)in_sizes; (void)n_in; (void)out_size; (void)ws_size;
    const float2* x   = (const float2*)d_in[0];   // complex64 [A,H,W]
    const float*  trj = (const float*)d_in[1];    // f32 [R,2,K]
    const float*  phi = (const float*)d_in[2];    // f32 [A,T]
    const float2* mps = (const float2*)d_in[3];   // complex64 [C,H,W]
    const float*  dcf = (const float*)d_in[4];    // f32 [R,K]
    const int*    sub = (const int*)d_in[5];      // i32 [T]
    float2* ws  = (float2*)d_ws;
    float2* out = (float2*)d_out;                 // complex64 [T,C,K] interleaved

    nudft_tables<<<(2 * R_ * K_ * 64) / 256, 256, 0, stream>>>(trj, ws);
    build_xs   <<<(AC_ * N_) / 256,        256, 0, stream>>>(x, mps, ws);
    nudft_wmma <<<R_ * KT_ * NSPLIT,       256, 0, stream>>>(ws);
    project_out<<<(T_ * C_ * K_) / 256,    256, 0, stream>>>(phi, dcf, sub, ws, out);
}